// GCNIDS_4028679323807
// MI455X (gfx1250) — compile-verified
//
#include <hip/hip_runtime.h>

// ---------------------------------------------------------------------------
// GCN (2x GCNConv + linear head) for MI455X / gfx1250.
//  - Projections use v_wmma_f32_16x16x4_f32 (fp32 GEMM, one 16x16 tile/wave32)
//  - Edge aggregation: one wave per edge, native f32 global atomics (L2-resident:
//    feature buffers are 25.6 MB each, well inside the 192 MB L2)
//  - Self-loops folded into the bias+ReLU pass: relu(G + P*dinv^2 + b)
// ---------------------------------------------------------------------------

typedef __attribute__((ext_vector_type(2))) float v2f;
typedef __attribute__((ext_vector_type(8))) float v8f;

typedef long long edge_t;  // edge_index is int64 in the reference

// ---------------- small elementwise kernels ----------------

__global__ void fill_f32(float* __restrict__ p, float v, int n) {
    int i = blockIdx.x * blockDim.x + threadIdx.x;
    if (i < n) p[i] = v;
}

__global__ void deg_count(const edge_t* __restrict__ dst, float* __restrict__ deg, int E) {
    int e = blockIdx.x * blockDim.x + threadIdx.x;
    if (e < E) unsafeAtomicAdd(&deg[dst[e]], 1.0f);  // lowers to global_atomic_add_f32
}

__global__ void rsqrt_inplace(float* __restrict__ d, int n) {
    int i = blockIdx.x * blockDim.x + threadIdx.x;
    if (i < n) d[i] = rsqrtf(d[i]);  // deg >= 1 (self-loop), no zero guard needed
}

// ---------------- fp32 WMMA projection: Out[M,64] = A[M,64] @ W[64,64] ------
// One wave32 computes one 16x16 output tile; 4 waves/block cover all 64 cols.

__global__ __launch_bounds__(128) void gemm64_wmma(const float* __restrict__ A,
                                                   const float* __restrict__ W,
                                                   float* __restrict__ Out,
                                                   int nrows) {
    const int lane = threadIdx.x & 31;
    const int wave = threadIdx.x >> 5;      // column tile: n0 = wave*16
    const int half = lane >> 4;             // 0 or 1
    const int ml   = lane & 15;
    const int m0   = blockIdx.x * 16;
    const int n0   = wave * 16;
    if (m0 >= nrows) return;                // full blocks only (nrows % 16 == 0)

    const float* __restrict__ arow = A + (size_t)(m0 + ml) * 64;
    v8f c = {};
#pragma unroll
    for (int k0 = 0; k0 < 64; k0 += 4) {
        const int k = k0 + 2 * half;        // this lane's K pair
        // A: 16x4 tile, lane holds A[ml][k], A[ml][k+1] (contiguous -> 64b load)
        v2f a = *(const v2f*)(arow + k);
        // B: 4x16 tile, lane holds W[k][n0+ml], W[k+1][n0+ml]
        v2f b;
        b.x = W[(size_t)k * 64 + n0 + ml];
        b.y = W[(size_t)(k + 1) * 64 + n0 + ml];
        // 8 args: (neg_a, A, neg_b, B, c_mod, C, reuse_a, reuse_b)
        c = __builtin_amdgcn_wmma_f32_16x16x4_f32(false, a, false, b,
                                                  (short)0, c, false, false);
    }
    // D layout: vgpr i, half h -> row (i + 8*h), col ml
    float* __restrict__ ocol = Out + (size_t)m0 * 64 + n0 + ml;
#pragma unroll
    for (int i = 0; i < 8; ++i)
        ocol[(size_t)(half * 8 + i) * 64] = c[i];
}

// ---------------- edge aggregation: G[dst] += P[src] * dinv[src]*dinv[dst] --
// One wave per edge; each lane handles 2 of the 64 features.

__global__ __launch_bounds__(256) void edge_aggregate(const edge_t* __restrict__ ei,
                                                      const float* __restrict__ dinv,
                                                      const float* __restrict__ P,
                                                      float* __restrict__ G,
                                                      int E) {
    const int e    = (int)((blockIdx.x * 256u + threadIdx.x) >> 5);
    const int lane = threadIdx.x & 31;
    if (e >= E) return;
    const edge_t s = ei[e];          // src (row 0)
    const edge_t d = ei[(size_t)E + e];  // dst (row 1)
    const float norm = dinv[s] * dinv[d];
    const float2 v = ((const float2*)(P + (size_t)s * 64))[lane];
    float* gd = G + (size_t)d * 64 + lane * 2;
    unsafeAtomicAdd(gd,     v.x * norm);
    unsafeAtomicAdd(gd + 1, v.y * norm);
}

// ---------------- bias + self-loop + ReLU (in place on G) ------------------
// h = relu(G + P * dinv[node]^2 + b[feat])

__global__ void bias_selfloop_relu(float* __restrict__ G,
                                   const float* __restrict__ P,
                                   const float* __restrict__ dinv,
                                   const float* __restrict__ b,
                                   int n64) {
    int i = blockIdx.x * blockDim.x + threadIdx.x;
    if (i >= n64) return;
    const int node = i >> 6;
    const int f    = i & 63;
    const float di = dinv[node];
    const float v  = G[i] + P[i] * (di * di) + b[f];
    G[i] = v > 0.0f ? v : 0.0f;
}

// ---------------- final head: out[n] = dot(H[n,:], Wout) + bout ------------
// One wave per node; wave32 shuffle reduction.

__global__ __launch_bounds__(256) void head_dot(const float* __restrict__ H,
                                                const float* __restrict__ Wout,
                                                const float* __restrict__ bout,
                                                float* __restrict__ out,
                                                int n) {
    const int node = (int)((blockIdx.x * 256u + threadIdx.x) >> 5);
    const int lane = threadIdx.x & 31;
    if (node >= n) return;
    const float* h = H + (size_t)node * 64;
    float s = h[lane] * Wout[lane] + h[lane + 32] * Wout[lane + 32];
#pragma unroll
    for (int off = 16; off > 0; off >>= 1)
        s += __shfl_down(s, off, 32);
    if (lane == 0) out[node] = s + bout[0];
}

// ---------------------------------------------------------------------------

extern "C" void kernel_launch(void* const* d_in, const int* in_sizes, int n_in,
                              void* d_out, int out_size, void* d_ws, size_t ws_size,
                              hipStream_t stream) {
    const float*  x    = (const float*)d_in[0];
    const edge_t* ei   = (const edge_t*)d_in[1];   // [2, E] row-major: src row then dst row
    const float*  W1   = (const float*)d_in[2];
    const float*  b1   = (const float*)d_in[3];
    const float*  W2   = (const float*)d_in[4];
    const float*  b2   = (const float*)d_in[5];
    const float*  Wout = (const float*)d_in[6];
    const float*  bout = (const float*)d_in[7];
    float*        out  = (float*)d_out;

    const int N = in_sizes[0] / 64;
    const int E = in_sizes[1] / 2;

    // workspace: dinv[N] | P[N*64] | G[N*64]
    char* ws = (char*)d_ws;
    size_t off = 0;
    auto carve = [&](size_t bytes) -> float* {
        float* p = (float*)(ws + off);
        off += (bytes + 255) & ~(size_t)255;
        return p;
    };
    float* dinv = carve((size_t)N * sizeof(float));
    float* P    = carve((size_t)N * 64 * sizeof(float));
    float* G    = carve((size_t)N * 64 * sizeof(float));

    const int T = 256;
    const int n64 = N * 64;
    const int edgeWaveBlocks = (int)(((size_t)E * 32 + T - 1) / T);
    const int nodeWaveBlocks = (int)(((size_t)N * 32 + T - 1) / T);

    // degrees (with self-loop) -> dinv = 1/sqrt(deg)
    fill_f32<<<(N + T - 1) / T, T, 0, stream>>>(dinv, 1.0f, N);
    deg_count<<<(E + T - 1) / T, T, 0, stream>>>(ei + E, dinv, E);
    rsqrt_inplace<<<(N + T - 1) / T, T, 0, stream>>>(dinv, N);

    // ---- layer 1 ----
    gemm64_wmma<<<N / 16, 128, 0, stream>>>(x, W1, P, N);
    fill_f32<<<(n64 + T - 1) / T, T, 0, stream>>>(G, 0.0f, n64);
    edge_aggregate<<<edgeWaveBlocks, T, 0, stream>>>(ei, dinv, P, G, E);
    bias_selfloop_relu<<<(n64 + T - 1) / T, T, 0, stream>>>(G, P, dinv, b1, n64);

    // ---- layer 2 ----
    gemm64_wmma<<<N / 16, 128, 0, stream>>>(G, W2, P, N);
    fill_f32<<<(n64 + T - 1) / T, T, 0, stream>>>(G, 0.0f, n64);
    edge_aggregate<<<edgeWaveBlocks, T, 0, stream>>>(ei, dinv, P, G, E);
    bias_selfloop_relu<<<(n64 + T - 1) / T, T, 0, stream>>>(G, P, dinv, b2, n64);

    // ---- head ----
    head_dot<<<nodeWaveBlocks, T, 0, stream>>>(G, Wout, bout, out, N);
}